// xformer_Conv3D_Aware_CrossAttention_55190329754002
// MI455X (gfx1250) — compile-verified
//
#include <hip/hip_runtime.h>
#include <hip/hip_bf16.h>

// ---------------------------------------------------------------------------
// Triplane cross-attention for MI455X (gfx1250, wave32, WMMA).
// Pipeline:
//   1) Q  = x  @ wq^T      (3072x768  @ 768x768 )  -- WMMA f32 GEMM
//   2) KV = x  @ w_kv^T    (3072x768  @ 768x1536)  -- WMMA f32 GEMM
//   3) attention (gathered 32-token context, softmax, weighted V) -- VALU
//   4) out = AO @ proj_w^T + proj_b (3072x768 @ 768x768) -- WMMA f32 GEMM
// ---------------------------------------------------------------------------

typedef float v2f __attribute__((ext_vector_type(2)));
typedef float v8f __attribute__((ext_vector_type(8)));

#define TILE_M 128
#define TILE_N 128
#define TILE_K 16
#define LDS_PAD 20   // floats per LDS row: 16 data + 4 pad (16B-aligned, conflict-free)

// C[m, o] = sum_k A[m, k] * W[o, k]  (+ bias[o])
// A: M x K row-major.  W: Nout x K row-major (i.e. B-matrix is W^T).
__global__ __launch_bounds__(256)
void gemm_nt_wmma(const float* __restrict__ A, int M, int K,
                  const float* __restrict__ W, int Nout,
                  const float* __restrict__ bias,
                  float* __restrict__ Cmat)
{
    __shared__ __align__(16) float As[TILE_M * LDS_PAD];
    __shared__ __align__(16) float Bs[TILE_N * LDS_PAD];

    const int tid   = threadIdx.x;
    const int lane  = tid & 31;
    const int wave  = tid >> 5;     // 0..7
    const int waveM = wave >> 2;    // 0..1  -> 64 rows each
    const int waveN = wave & 3;     // 0..3  -> 32 cols each

    const int m0 = blockIdx.y * TILE_M;
    const int o0 = blockIdx.x * TILE_N;

    const int lm = lane & 15;             // M (or N) index within 16
    const int lk = (lane >> 4) << 1;      // K sub-offset: 0 or 2 (ISA A/B layout)

    v8f acc[4][2];
#pragma unroll
    for (int i = 0; i < 4; ++i)
#pragma unroll
        for (int j = 0; j < 2; ++j)
#pragma unroll
            for (int r = 0; r < 8; ++r)
                acc[i][j][r] = 0.0f;

    for (int k0 = 0; k0 < K; k0 += TILE_K) {
        // ---- cooperative tile load: 512 float4 per matrix, 2 per thread ----
#pragma unroll
        for (int rep = 0; rep < 2; ++rep) {
            const int idx = tid + rep * 256;   // 0..511
            const int row = idx >> 2;          // 0..127
            const int kc  = (idx & 3) << 2;    // 0,4,8,12
            const float4 av = *(const float4*)(A + (size_t)(m0 + row) * K + k0 + kc);
            *(float4*)(&As[row * LDS_PAD + kc]) = av;
            const float4 bv = *(const float4*)(W + (size_t)(o0 + row) * K + k0 + kc);
            *(float4*)(&Bs[row * LDS_PAD + kc]) = bv;
        }
        __syncthreads();

        // ---- 4 k-substeps of V_WMMA_F32_16X16X4_F32 ----
#pragma unroll
        for (int ks = 0; ks < 4; ++ks) {
            const int kof = ks * 4 + lk;
            v2f af[4];
            v2f bf[2];
#pragma unroll
            for (int i = 0; i < 4; ++i) {
                const float* p = &As[(waveM * 64 + i * 16 + lm) * LDS_PAD + kof];
                af[i].x = p[0];
                af[i].y = p[1];
            }
#pragma unroll
            for (int j = 0; j < 2; ++j) {
                const float* p = &Bs[(waveN * 32 + j * 16 + lm) * LDS_PAD + kof];
                bf[j].x = p[0];
                bf[j].y = p[1];
            }
#pragma unroll
            for (int i = 0; i < 4; ++i)
#pragma unroll
                for (int j = 0; j < 2; ++j)
                    acc[i][j] = __builtin_amdgcn_wmma_f32_16x16x4_f32(
                        /*neg_a=*/false, af[i],
                        /*neg_b=*/false, bf[j],
                        /*c_mod=*/(short)0, acc[i][j],
                        /*reuse_a=*/false, /*reuse_b=*/false);
        }
        __syncthreads();
    }

    // ---- write back (C layout: VGPR r -> M = r + 8*(lane>=16), N = lane&15) ----
    const int rhi  = (lane >> 4) * 8;
    const int colL = lane & 15;
#pragma unroll
    for (int i = 0; i < 4; ++i)
#pragma unroll
        for (int j = 0; j < 2; ++j) {
            const int o = o0 + waveN * 32 + j * 16 + colL;
            const float badd = (bias != nullptr) ? bias[o] : 0.0f;
            const int mb = m0 + waveM * 64 + i * 16 + rhi;
#pragma unroll
            for (int r = 0; r < 8; ++r)
                Cmat[(size_t)(mb + r) * Nout + o] = acc[i][j][r] + badd;
        }
}

// ---------------------------------------------------------------------------
// Attention: one block per (g, b, row a); one thread per (query col q, head h).
// Context: 16 row-ctx tokens from plane (g+1)%3 (row a) followed by 16 col-ctx
// tokens from plane (g+2)%3 (col q). Output rows written in (g,b,n) flat order.
// ---------------------------------------------------------------------------
__global__ __launch_bounds__(192)
void triplane_attn(const float* __restrict__ Qb,   // rows in (b,g,n) order, 768 wide
                   const float* __restrict__ KVb,  // rows in (b,g,n) order, 1536 wide
                   float* __restrict__ AO)         // rows in (g,b,n) order, 768 wide
{
    const int G = 3, Bn = 4, P = 16, C = 768, H = 12, D = 64;

    const int blk = blockIdx.x;           // 0..191
    const int g = blk / (Bn * P);
    const int b = (blk / P) % Bn;
    const int a = blk % P;

    const int tid = threadIdx.x;          // 0..191  (q-major for coalescing)
    const int q = tid / H;                // query column 0..15
    const int h = tid % H;                // head 0..11

    const int n  = a * P + q;
    const int g1 = (g + 1) % G;
    const int g2 = (g + 2) % G;

    // q vector for this head (64 floats) in registers
    const float4* qp = (const float4*)(Qb + ((size_t)((b * G + g) * 256 + n)) * C + h * D);
    float4 qr[16];
#pragma unroll
    for (int i = 0; i < 16; ++i) qr[i] = qp[i];

    const size_t rowRowBase = (size_t)((b * G + g1) * 256 + a * P);   // + j
    const size_t colRowBase = (size_t)((b * G + g2) * 256 + q);       // + j*16

    float s[32];
#pragma unroll
    for (int j = 0; j < 16; ++j) {
        const float4* kp = (const float4*)(KVb + (rowRowBase + j) * 1536 + h * D);
        float acc = 0.0f;
#pragma unroll
        for (int i = 0; i < 16; ++i) {
            const float4 kv4 = kp[i];
            acc += qr[i].x * kv4.x + qr[i].y * kv4.y + qr[i].z * kv4.z + qr[i].w * kv4.w;
        }
        s[j] = acc * 0.125f;   // 1/sqrt(64)
    }
#pragma unroll
    for (int j = 0; j < 16; ++j) {
        const float4* kp = (const float4*)(KVb + (colRowBase + (size_t)j * 16) * 1536 + h * D);
        float acc = 0.0f;
#pragma unroll
        for (int i = 0; i < 16; ++i) {
            const float4 kv4 = kp[i];
            acc += qr[i].x * kv4.x + qr[i].y * kv4.y + qr[i].z * kv4.z + qr[i].w * kv4.w;
        }
        s[16 + j] = acc * 0.125f;
    }

    // softmax over 32 context slots
    float mx = s[0];
#pragma unroll
    for (int j = 1; j < 32; ++j) mx = fmaxf(mx, s[j]);
    float sum = 0.0f;
#pragma unroll
    for (int j = 0; j < 32; ++j) { s[j] = __expf(s[j] - mx); sum += s[j]; }
    const float inv = 1.0f / sum;

    float4 o[16];
#pragma unroll
    for (int i = 0; i < 16; ++i) { o[i].x = 0.f; o[i].y = 0.f; o[i].z = 0.f; o[i].w = 0.f; }

#pragma unroll
    for (int j = 0; j < 32; ++j) {
        const size_t row = (j < 16) ? (rowRowBase + j)
                                    : (colRowBase + (size_t)(j - 16) * 16);
        const float4* vp = (const float4*)(KVb + row * 1536 + 768 + h * D);
        const float w = s[j] * inv;
#pragma unroll
        for (int i = 0; i < 16; ++i) {
            const float4 v4 = vp[i];
            o[i].x += w * v4.x; o[i].y += w * v4.y; o[i].z += w * v4.z; o[i].w += w * v4.w;
        }
    }

    float4* op = (float4*)(AO + ((size_t)((g * Bn + b) * 256 + n)) * C + h * D);
#pragma unroll
    for (int i = 0; i < 16; ++i) op[i] = o[i];
}

// ---------------------------------------------------------------------------

extern "C" void kernel_launch(void* const* d_in, const int* in_sizes, int n_in,
                              void* d_out, int out_size, void* d_ws, size_t ws_size,
                              hipStream_t stream) {
    const float* x      = (const float*)d_in[0];   // (B,G,N,C) = (4,3,256,768)
    const float* wq     = (const float*)d_in[1];   // (768,768)
    const float* w_kv   = (const float*)d_in[2];   // (1536,768)
    const float* proj_w = (const float*)d_in[3];   // (768,768)
    const float* proj_b = (const float*)d_in[4];   // (768,)
    float* out = (float*)d_out;

    const int M = 3072;   // B*G*N tokens
    const int C = 768;

    float* Qb  = (float*)d_ws;                       // 3072 x 768
    float* KVb = Qb  + (size_t)M * C;                // 3072 x 1536
    float* AO  = KVb + (size_t)M * 2 * C;            // 3072 x 768

    dim3 blk(256);
    // 1) Q = x @ wq^T
    gemm_nt_wmma<<<dim3(C / TILE_N, M / TILE_M), blk, 0, stream>>>(
        x, M, C, wq, C, nullptr, Qb);
    // 2) KV = x @ w_kv^T
    gemm_nt_wmma<<<dim3(2 * C / TILE_N, M / TILE_M), blk, 0, stream>>>(
        x, M, C, w_kv, 2 * C, nullptr, KVb);
    // 3) attention
    triplane_attn<<<dim3(192), dim3(192), 0, stream>>>(Qb, KVb, AO);
    // 4) out = AO @ proj_w^T + proj_b
    gemm_nt_wmma<<<dim3(C / TILE_N, M / TILE_M), blk, 0, stream>>>(
        AO, M, C, proj_w, C, proj_b, out);
}